// BNetEncDec_29102698397992
// MI455X (gfx1250) — compile-verified
//
#include <hip/hip_runtime.h>
#include <math.h>

typedef float v2f __attribute__((ext_vector_type(2)));
typedef float v8f __attribute__((ext_vector_type(8)));

#define LATD 64
#define LDP  66                 // even row stride: 8B-aligned A pairs + conflict-free A loads
#define MSZ  (LATD * LDP)       // floats per 64x64 LDS matrix
#define NTHR 256                // 8 waves of 32
#define VECOFF (5 * MSZ)
#define REDOFF (VECOFF + 512)
#define LDS_FLOATS (REDOFF + 512)

// D = scale * X @ Y'   where Y' = (FUSE ? cdiag*I - Y : Y), all 64x64 in LDS.
// Wave w owns output tiles (2w, 2w+1) of the 4x4 16x16-tile grid; both tiles
// share the same tile ROW, so the A-fragment is loaded once (ds_load_b64) and
// feeds two v_wmma_f32_16x16x4_f32 chains (two accumulators).
// fp32 WMMA 16x16x4 layout: A-frag lane = {M=lane%16, K=(lane/16)*2 .. +1},
// B-frag lane = {K=(lane/16)*2 .. +1, N=lane%16}, C/D: vgpr r -> row r + 8*(lane/16).
template <bool FUSE>
__device__ __forceinline__ void mm64(const float* X, const float* Y,
                                     float cdiag, float scale, float* D,
                                     int lane, int wave) {
  const int m16 = lane & 15;
  const int hi  = lane >> 4;            // 0 or 1
  const int ti  = (wave >> 1) << 4;     // tile row (shared by both tiles)
  const int tj0 = (wave & 1) << 5;      // first tile col; second is +16
  const int n0  = tj0 + m16;
  const int n1  = n0 + 16;

  v8f acc0 = {0.f, 0.f, 0.f, 0.f, 0.f, 0.f, 0.f, 0.f};
  v8f acc1 = {0.f, 0.f, 0.f, 0.f, 0.f, 0.f, 0.f, 0.f};

  #pragma unroll
  for (int k = 0; k < 64; k += 4) {
    const int kk = k + hi * 2;
    // A pair: (row*66 + even)*4 is 8-byte aligned -> single ds_load_b64
    const v2f a = *(const v2f*)&X[(ti + m16) * LDP + kk];
    float y00 = Y[kk * LDP + n0];
    float y01 = Y[(kk + 1) * LDP + n0];
    float y10 = Y[kk * LDP + n1];
    float y11 = Y[(kk + 1) * LDP + n1];
    if (FUSE) {
      y00 = ((kk     == n0) ? cdiag : 0.0f) - y00;
      y01 = ((kk + 1 == n0) ? cdiag : 0.0f) - y01;
      y10 = ((kk     == n1) ? cdiag : 0.0f) - y10;
      y11 = ((kk + 1 == n1) ? cdiag : 0.0f) - y11;
    }
    v2f b0; b0.x = y00; b0.y = y01;
    v2f b1; b1.x = y10; b1.y = y11;
    acc0 = __builtin_amdgcn_wmma_f32_16x16x4_f32(
        false, a, false, b0, (short)0, acc0, false, false);
    acc1 = __builtin_amdgcn_wmma_f32_16x16x4_f32(
        false, a, false, b1, (short)0, acc1, false, false);
  }
  #pragma unroll
  for (int r = 0; r < 8; ++r) {
    D[(ti + hi * 8 + r) * LDP + n0] = acc0[r] * scale;
    D[(ti + hi * 8 + r) * LDP + n1] = acc1[r] * scale;
  }
}

__global__ void __launch_bounds__(NTHR)
bnet_cayley_kernel(const float* __restrict__ li_lat, const float* __restrict__ gi_lat,
                   const float* __restrict__ Wup, const float* __restrict__ bup,
                   const float* __restrict__ W1, const float* __restrict__ b1,
                   const float* __restrict__ W2, const float* __restrict__ b2,
                   const float* __restrict__ W3, const float* __restrict__ b3,
                   const float* __restrict__ W4, const float* __restrict__ b4,
                   const float* __restrict__ W5, const float* __restrict__ b5,
                   const float* __restrict__ W6, const float* __restrict__ b6,
                   float* __restrict__ out_recons, float* __restrict__ out_pred,
                   float* __restrict__ out_latent) {
  extern __shared__ float lds[];
  float* B   = lds;            // I - A (persistent); note Q = I + A = 2I - B
  float* Za  = lds + 1 * MSZ;
  float* Zb  = lds + 2 * MSZ;
  float* M1  = lds + 3 * MSZ;
  float* M2  = lds + 4 * MSZ;
  float* gl  = lds + VECOFF;   // gi_latent row
  float* liS = gl + 64;
  float* mis = liS + 64;
  float* hv  = mis + 64;
  float* fc2 = hv + 64;
  float* lat = fc2 + 64;
  float* t3  = lat + 64;
  float* t5  = t3 + 64;
  float* red  = lds + REDOFF;
  float* redM = red + 256;

  const int n    = blockIdx.x;
  const int tid  = threadIdx.x;
  const int lane = tid & 31;
  const int wave = tid >> 5;

  if (tid < 64) {
    gl[tid]  = gi_lat[n * 64 + tid];
    liS[tid] = li_lat[n * 64 + tid];
    B[tid * LDP + tid] = 1.0f;   // diag of B = I - A
  }
  __syncthreads();

  // gi = gi_latent @ Wup^T + bup, scattered as B = I - (A_raw^T - A_raw).
  // For lower-tri pair (r>c), e = gi[idx]:  B[r][c] = +e, B[c][r] = -e.
  for (int p = tid; p < 2016; p += NTHR) {
    int r = (int)((1.0f + sqrtf(1.0f + 8.0f * (float)p)) * 0.5f);
    while (r * (r - 1) / 2 > p) --r;
    while ((r + 1) * r / 2 <= p) ++r;
    const int c = p - r * (r - 1) / 2;
    const float* wrow = Wup + p * 64;
    float acc = bup[p];
    #pragma unroll 8
    for (int k = 0; k < 64; ++k) acc += wrow[k] * gl[k];
    B[r * LDP + c] =  acc;
    B[c * LDP + r] = -acc;
  }
  __syncthreads();

  // norms = sum(|B|) * max(|B|);  Z0 = B^T / norms
  float s = 0.f, mx = 0.f;
  for (int idx = tid; idx < 4096; idx += NTHR) {
    const float v = fabsf(B[(idx >> 6) * LDP + (idx & 63)]);
    s += v; mx = fmaxf(mx, v);
  }
  red[tid] = s; redM[tid] = mx;
  __syncthreads();
  for (int st = NTHR / 2; st > 0; st >>= 1) {
    if (tid < st) {
      red[tid] += red[tid + st];
      redM[tid] = fmaxf(redM[tid], redM[tid + st]);
    }
    __syncthreads();
  }
  const float inv_norm = 1.0f / (red[0] * redM[0]);
  for (int idx = tid; idx < 4096; idx += NTHR) {
    const int i = idx >> 6, j = idx & 63;
    Za[i * LDP + j] = B[j * LDP + i] * inv_norm;
  }
  __syncthreads();

  // 10 Newton-Schulz steps: Z = 0.25 * Z @ (13I - BZ@(15I - BZ@(7I - BZ)))
  float* Zc = Za;
  float* Zo = Zb;
  for (int it = 0; it < 10; ++it) {
    mm64<false>(B,  Zc, 0.0f,  1.0f,  M1, lane, wave); __syncthreads(); // M1 = B@Z
    mm64<true >(M1, M1, 7.0f,  1.0f,  Zo, lane, wave); __syncthreads(); // Zo = BZ@(7I-BZ)
    mm64<true >(M1, Zo, 15.0f, 1.0f,  M2, lane, wave); __syncthreads(); // M2 = BZ@(15I-Zo)
    mm64<true >(Zc, M2, 13.0f, 0.25f, Zo, lane, wave); __syncthreads(); // Znew
    float* t = Zc; Zc = Zo; Zo = t;
  }
  // R = Z @ Q = Z @ (2I - B)  -> M1
  mm64<true>(Zc, B, 2.0f, 1.0f, M1, lane, wave);
  __syncthreads();
  const float* R = M1;

  // mis = R^T @ li
  if (tid < 64) {
    float a = 0.f;
    for (int i = 0; i < 64; ++i) a += R[i * LDP + tid] * liS[i];
    mis[tid] = a;
  }
  __syncthreads();
  if (tid < 64) {                       // h = relu(mis @ W1^T + b1)
    float a = b1[tid];
    const float* w = W1 + tid * 64;
    for (int k = 0; k < 64; ++k) a += w[k] * mis[k];
    hv[tid] = fmaxf(a, 0.f);
  }
  __syncthreads();
  if (tid < 64) {                       // fc2 = h @ W2^T + b2
    float a = b2[tid];
    const float* w = W2 + tid * 64;
    for (int k = 0; k < 64; ++k) a += w[k] * hv[k];
    fc2[tid] = a;
  }
  __syncthreads();
  if (tid < 64) {                       // latent = R @ fc2
    float a = 0.f;
    for (int j = 0; j < 64; ++j) a += R[tid * LDP + j] * fc2[j];
    lat[tid] = a;
    out_latent[n * 64 + tid] = a;
  }
  __syncthreads();
  if (tid < 64) {
    float a3 = b3[tid], a5 = b5[tid];
    const float* w3 = W3 + tid * 64;
    const float* w5 = W5 + tid * 64;
    for (int k = 0; k < 64; ++k) { a3 += w3[k] * lat[k]; a5 += w5[k] * lat[k]; }
    t3[tid] = fmaxf(a3, 0.f);
    t5[tid] = fmaxf(a5, 0.f);
  }
  __syncthreads();
  if (tid < 64) {                       // recons = t3 @ W4^T + b4
    float a = b4[tid];
    const float* w = W4 + tid * 64;
    for (int k = 0; k < 64; ++k) a += w[k] * t3[k];
    out_recons[n * 64 + tid] = a;
  }
  if (tid < 2) {                        // pred = t5 @ W6^T + b6
    float a = b6[tid];
    const float* w = W6 + tid * 64;
    for (int k = 0; k < 64; ++k) a += w[k] * t5[k];
    out_pred[n * 2 + tid] = a;
  }
}

extern "C" void kernel_launch(void* const* d_in, const int* in_sizes, int n_in,
                              void* d_out, int out_size, void* d_ws, size_t ws_size,
                              hipStream_t stream) {
  (void)n_in; (void)d_ws; (void)ws_size; (void)out_size;
  const float* li  = (const float*)d_in[0];
  const float* gi  = (const float*)d_in[1];
  const float* Wup = (const float*)d_in[2];
  const float* bup = (const float*)d_in[3];
  const float* W1  = (const float*)d_in[4];
  const float* b1  = (const float*)d_in[5];
  const float* W2  = (const float*)d_in[6];
  const float* b2  = (const float*)d_in[7];
  const float* W3  = (const float*)d_in[8];
  const float* b3  = (const float*)d_in[9];
  const float* W4  = (const float*)d_in[10];
  const float* b4  = (const float*)d_in[11];
  const float* W5  = (const float*)d_in[12];
  const float* b5  = (const float*)d_in[13];
  const float* W6  = (const float*)d_in[14];
  const float* b6  = (const float*)d_in[15];

  const int N = in_sizes[0] / 64;       // 4096 samples
  float* out        = (float*)d_out;    // (recons, pred, latent) concatenated
  float* out_recons = out;
  float* out_pred   = out + (size_t)N * 64;
  float* out_latent = out + (size_t)N * 66;

  const size_t shmem = (size_t)LDS_FLOATS * sizeof(float);  // ~88.6 KB dynamic LDS
  bnet_cayley_kernel<<<N, NTHR, shmem, stream>>>(
      li, gi, Wup, bup, W1, b1, W2, b2, W3, b3, W4, b4, W5, b5, W6, b6,
      out_recons, out_pred, out_latent);
}